// S4Layer_43688407335399
// MI455X (gfx1250) — compile-verified
//
#include <hip/hip_runtime.h>

// ---------------------------------------------------------------------------
// S4 layer forward for MI455X (gfx1250, wave32).
//   K-kernel construction (Cauchy + FFT) and FFT convolution, with every
//   length-4096 transform done as three radix-16 stages whose DFT-16 cores
//   are 16x16x16 complex matmuls on V_WMMA_F32_16X16X4_F32.
//   All twiddles come from a 4096-entry LDS LUT (exact grid angles) instead
//   of per-use sincosf, eliminating the libm slow-path VALU/branch bloat.
// ---------------------------------------------------------------------------

#define TWO_PI 6.28318530717958647692f

typedef __attribute__((ext_vector_type(2))) float v2f;
typedef __attribute__((ext_vector_type(8))) float v8f;

#if defined(__HIP_DEVICE_COMPILE__) && !__has_builtin(__builtin_amdgcn_wmma_f32_16x16x4_f32)
#error "gfx1250 device builtin __builtin_amdgcn_wmma_f32_16x16x4_f32 not available"
#endif

// cs/sn: cos/sin(2*pi*i/4096), i in [0,4096). c16/s16: DFT-16 matrix tables.
__device__ __forceinline__ void build_tables(float* cs, float* sn,
                                             float* c16, float* s16) {
  const int tid = threadIdx.x;
#pragma unroll
  for (int s = 0; s < 16; ++s) {
    const int i = tid + 256 * s;
    float sv, cv;
    sincosf(TWO_PI * (float)i / 4096.0f, &sv, &cv);
    cs[i] = cv;
    sn[i] = sv;
  }
  if (tid < 256) {
    const int u = tid >> 4, t = tid & 15;
    float sv, cv;
    sincosf(TWO_PI * (float)(u * t) / 16.0f, &sv, &cv);
    c16[tid] = cv;
    s16[tid] = sv;
  }
}

// One radix-16 stage over a 4096-point LDS-resident signal.
// M = stage span (4096, 256, 16). 256 DFT-16 instances per stage are grouped
// into 16 column-tiles of 16; each of the 8 waves owns 2 tiles and performs a
// full 16x16x16 complex matmul as 16 chained V_WMMA_F32_16X16X4_F32 ops.
// SIGMA=+1: forward DFT (twiddle e^{-2pi i j u / M} applied on outputs).
// SIGMA=-1: inverse stage (conj twiddle e^{+2pi i j t / M} applied on inputs,
//           IDFT-16 matmul; the missing 1/16 per stage -> 1/4096 at the end).
template <int M, bool IN_TW, bool OUT_TW, int SIGMA>
__device__ __forceinline__ void fft_stage16(float* __restrict__ re,
                                            float* __restrict__ im,
                                            const float* __restrict__ c16,
                                            const float* __restrict__ s16,
                                            const float* __restrict__ cs,
                                            const float* __restrict__ sn) {
  constexpr int M16 = M / 16;
  constexpr int LSTRIDE = 4096 / M;   // twiddle-LUT step for this stage
  const int tid  = threadIdx.x;
  const int lane = tid & 31;
  const int wave = tid >> 5;
  const int nlo  = lane & 15;
  const int khi  = (lane >> 4) << 1;  // lanes 16..31 carry K+2 / K+3 (ISA 7.12.2)
  const float asgn = -(float)SIGMA;   // A_imag sign and twiddle sin sign

#pragma unroll
  for (int tile = 0; tile < 2; ++tile) {
    const int inst = (wave + 8 * tile) * 16 + nlo;  // DFT instance (column)
    const int blk  = inst / M16;
    const int j    = inst - blk * M16;
    const int base = blk * M + j;

    v8f accR = {};
    v8f accI = {};
#pragma unroll
    for (int kb = 0; kb < 16; kb += 4) {
      const int k0 = kb + khi;
      const int k1 = k0 + 1;
      // A fragments: W[u][t] = cos + i*asgn*sin, rows u = lane%16.
      v2f ac, ai, ain;
      ac.x  = c16[nlo * 16 + k0];
      ac.y  = c16[nlo * 16 + k1];
      ai.x  = asgn * s16[nlo * 16 + k0];
      ai.y  = asgn * s16[nlo * 16 + k1];
      ain.x = -ai.x;
      ain.y = -ai.y;
      // B fragments: element t of this instance, gathered from LDS.
      float xr0 = re[base + k0 * M16], xi0 = im[base + k0 * M16];
      float xr1 = re[base + k1 * M16], xi1 = im[base + k1 * M16];
      if (IN_TW) {
        const int i0 = (j * k0 * LSTRIDE) & 4095;
        const int i1 = (j * k1 * LSTRIDE) & 4095;
        const float c0 = cs[i0], s0 = asgn * sn[i0];
        const float c1 = cs[i1], s1 = asgn * sn[i1];
        float t0 = xr0 * c0 - xi0 * s0; xi0 = xr0 * s0 + xi0 * c0; xr0 = t0;
        float t1 = xr1 * c1 - xi1 * s1; xi1 = xr1 * s1 + xi1 * c1; xr1 = t1;
      }
      v2f br, bi;
      br.x = xr0; br.y = xr1;
      bi.x = xi0; bi.y = xi1;

      // Dr += Ar*Br - Ai*Bi ; Di += Ar*Bi + Ai*Br
      accR = __builtin_amdgcn_wmma_f32_16x16x4_f32(false, ac,  false, br, (short)0, accR, false, false);
      accR = __builtin_amdgcn_wmma_f32_16x16x4_f32(false, ain, false, bi, (short)0, accR, false, false);
      accI = __builtin_amdgcn_wmma_f32_16x16x4_f32(false, ac,  false, bi, (short)0, accI, false, false);
      accI = __builtin_amdgcn_wmma_f32_16x16x4_f32(false, ai,  false, br, (short)0, accI, false, false);
    }

#pragma unroll
    for (int r = 0; r < 8; ++r) {
      const int u = r + ((lane >> 4) << 3);  // C/D rows r and r+8 per VGPR
      float yr = accR[r], yi = accI[r];
      if (OUT_TW) {
        const int iw = (j * u * LSTRIDE) & 4095;
        const float c = cs[iw], s = asgn * sn[iw];
        float t = yr * c - yi * s; yi = yr * s + yi * c; yr = t;
      }
      re[base + u * M16] = yr;
      im[base + u * M16] = yi;
    }
  }
}

__device__ __forceinline__ void fft4096_fwd(float* re, float* im,
                                            const float* c16, const float* s16,
                                            const float* cs, const float* sn) {
  fft_stage16<4096, false, true,  1>(re, im, c16, s16, cs, sn); __syncthreads();
  fft_stage16<256,  false, true,  1>(re, im, c16, s16, cs, sn); __syncthreads();
  fft_stage16<16,   false, false, 1>(re, im, c16, s16, cs, sn); __syncthreads();
}

__device__ __forceinline__ void fft4096_inv(float* re, float* im,
                                            const float* c16, const float* s16,
                                            const float* cs, const float* sn) {
  fft_stage16<16,   false, false, -1>(re, im, c16, s16, cs, sn); __syncthreads();
  fft_stage16<256,  true,  false, -1>(re, im, c16, s16, cs, sn); __syncthreads();
  fft_stage16<4096, true,  false, -1>(re, im, c16, s16, cs, sn); __syncthreads();
}

// ---------------------------------------------------------------------------
// Kernel A: per-channel SSM kernel spectrum Khat[d, 0..4096) (digit-rev order).
//   at_roots (Cauchy over N=64) -> K[t] = real(FFT4096(even-interleave))/2048
//   -> Khat = FFT4096(pad(K)).
// ---------------------------------------------------------------------------
__global__ __launch_bounds__(256) void s4_kernel_K(
    const float* __restrict__ Bm, const float* __restrict__ Ctm,
    const float* __restrict__ log_step,
    const float2* __restrict__ p, const float2* __restrict__ q,
    const float2* __restrict__ lam, float2* __restrict__ Khat) {
  __shared__ float reb[4096], imb[4096];
  __shared__ float cs[4096], sn[4096];
  __shared__ float c16[256], s16[256];
  __shared__ float2 lamS[64], v01S[64], v10S[64], v11S[64];
  __shared__ float v00S[64];

  const int d = blockIdx.x;
  const int tid = threadIdx.x;
  build_tables(cs, sn, c16, s16);
  if (tid < 64) {
    const int n = tid;
    const float Bn = Bm[d * 64 + n], Cn = Ctm[d * 64 + n];
    const float2 pn = p[n], qn = q[n];
    lamS[n] = lam[n];
    v00S[n] = Cn * Bn;                                             // Ct*B (real)
    v01S[n] = make_float2(Cn * pn.x, Cn * pn.y);                   // Ct*p
    v10S[n] = make_float2(qn.x * Bn, -qn.y * Bn);                  // conj(q)*B
    v11S[n] = make_float2(qn.x * pn.x + qn.y * pn.y,
                          qn.x * pn.y - qn.y * pn.x);              // conj(q)*p
  }
  const float g0 = 2.0f / expf(log_step[d]);
  __syncthreads();

  for (int s = 0; s < 8; ++s) {
    const int l = tid + 256 * s;
    const int wl = (2 * l) & 4095;           // omega_l = e^{2 pi i l / 2048}
    const float wr_ = cs[wl], wi_ = sn[wl];
    const float denr = 1.0f + wr_, deni = wi_;
    const float dd = denr * denr + deni * deni;
    const float numr = 1.0f - wr_, numi = -wi_;
    const float gr = g0 * (numr * denr + numi * deni) / dd;
    const float gi = g0 * (numi * denr - numr * deni) / dd;
    const float cr = 2.0f * denr / dd, ci = -2.0f * deni / dd;  // c = 2/(1+w)
    float k00r = 0, k00i = 0, k01r = 0, k01i = 0;
    float k10r = 0, k10i = 0, k11r = 0, k11i = 0;
    for (int n = 0; n < 64; ++n) {
      const float2 ln = lamS[n];
      const float drr = gr - ln.x, dii = gi - ln.y;
      const float inv = 1.0f / (drr * drr + dii * dii);
      const float rr = drr * inv, ri = -dii * inv;  // 1/(g - lam)
      k00r += v00S[n] * rr; k00i += v00S[n] * ri;
      { const float2 v = v01S[n]; k01r += v.x * rr - v.y * ri; k01i += v.x * ri + v.y * rr; }
      { const float2 v = v10S[n]; k10r += v.x * rr - v.y * ri; k10i += v.x * ri + v.y * rr; }
      { const float2 v = v11S[n]; k11r += v.x * rr - v.y * ri; k11i += v.x * ri + v.y * rr; }
    }
    const float er = 1.0f + k11r, ei = k11i;
    const float ed = er * er + ei * ei;
    const float tr = k01r * k10r - k01i * k10i, ti = k01r * k10i + k01i * k10r;
    const float qr = (tr * er + ti * ei) / ed, qi = (ti * er - tr * ei) / ed;
    const float ar = k00r - qr, ai = k00i - qi;
    reb[2 * l]     = ar * cr - ai * ci;   // even-interleave: z[2l] = at_roots[l]
    imb[2 * l]     = ar * ci + ai * cr;
    reb[2 * l + 1] = 0.0f;
    imb[2 * l + 1] = 0.0f;
  }
  __syncthreads();
  fft4096_fwd(reb, imb, c16, s16, cs, sn);
  // K time-domain, natural order: base-16 3-digit reversal (an involution).
  for (int s = 0; s < 16; ++s) {
    const int i = tid + 256 * s;
    const int r = ((i & 15) << 8) | (i & 0xF0) | (i >> 8);
    if (i < r) { const float t = reb[i]; reb[i] = reb[r]; reb[r] = t; }
  }
  __syncthreads();
  for (int s = 0; s < 16; ++s) {
    const int i = tid + 256 * s;
    reb[i] = (i < 2048) ? reb[i] * (1.0f / 2048.0f) : 0.0f;  // real K, zero-pad
    imb[i] = 0.0f;
  }
  __syncthreads();
  fft4096_fwd(reb, imb, c16, s16, cs, sn);
  for (int s = 0; s < 16; ++s) {
    const int i = tid + 256 * s;
    Khat[(size_t)d * 4096 + i] = make_float2(reb[i], imb[i]);
  }
}

// ---------------------------------------------------------------------------
// Tiled transposes so all FFT-row traffic is fully coalesced.
// ---------------------------------------------------------------------------
__global__ __launch_bounds__(256) void s4_tin(const float* __restrict__ u,
                                              float* __restrict__ ut) {
  __shared__ float tile[32][33];
  const int b = blockIdx.z, t0 = blockIdx.x * 32, d0 = blockIdx.y * 32;
  const int lx = threadIdx.x & 31, ly = threadIdx.x >> 5;
#pragma unroll
  for (int r = 0; r < 32; r += 8)
    tile[ly + r][lx] = u[((size_t)b * 2048 + (t0 + ly + r)) * 512 + (d0 + lx)];
  __syncthreads();
#pragma unroll
  for (int r = 0; r < 32; r += 8)
    ut[((size_t)b * 512 + (d0 + ly + r)) * 2048 + (t0 + lx)] = tile[lx][ly + r];
}

__global__ __launch_bounds__(256) void s4_tout(const float* __restrict__ yt,
                                               float* __restrict__ out) {
  __shared__ float tile[32][33];
  const int b = blockIdx.z, t0 = blockIdx.x * 32, d0 = blockIdx.y * 32;
  const int lx = threadIdx.x & 31, ly = threadIdx.x >> 5;
#pragma unroll
  for (int r = 0; r < 32; r += 8)
    tile[ly + r][lx] = yt[((size_t)b * 512 + (d0 + ly + r)) * 2048 + (t0 + lx)];
  __syncthreads();
#pragma unroll
  for (int r = 0; r < 32; r += 8)
    out[((size_t)b * 2048 + (t0 + ly + r)) * 512 + (d0 + lx)] = tile[lx][ly + r];
}

// ---------------------------------------------------------------------------
// Kernel B: per-(b,d) FFT convolution, in place over ut rows.
// ---------------------------------------------------------------------------
__global__ __launch_bounds__(256) void s4_conv(float* __restrict__ ut,
                                               const float2* __restrict__ Khat,
                                               const float* __restrict__ Dv) {
  __shared__ float reb[4096], imb[4096];
  __shared__ float cs[4096], sn[4096];
  __shared__ float uorig[2048];
  __shared__ float c16[256], s16[256];
  const int d = blockIdx.x & 511;
  const int b = blockIdx.x >> 9;
  const int tid = threadIdx.x;
  build_tables(cs, sn, c16, s16);
  float* __restrict__ urow = ut + ((size_t)b * 512 + d) * 2048;
  const float2* __restrict__ krow = Khat + (size_t)d * 4096;
#pragma unroll
  for (int s = 0; s < 16; ++s) __builtin_prefetch(&krow[tid + 256 * s], 0, 3);
  for (int s = 0; s < 8; ++s) {
    const int t = tid + 256 * s;
    const float v = urow[t];
    uorig[t] = v;
    reb[t] = v;           imb[t] = 0.0f;
    reb[t + 2048] = 0.0f; imb[t + 2048] = 0.0f;
  }
  __syncthreads();
  fft4096_fwd(reb, imb, c16, s16, cs, sn);
  for (int s = 0; s < 16; ++s) {  // pointwise * Khat (same digit-rev order)
    const int i = tid + 256 * s;
    const float2 kh = krow[i];
    const float xr = reb[i], xi = imb[i];
    reb[i] = xr * kh.x - xi * kh.y;
    imb[i] = xr * kh.y + xi * kh.x;
  }
  __syncthreads();
  fft4096_inv(reb, imb, c16, s16, cs, sn);
  const float Dd = Dv[d];
  for (int s = 0; s < 8; ++s) {
    const int t = tid + 256 * s;
    urow[t] = reb[t] * (1.0f / 4096.0f) + Dd * uorig[t];
  }
}

// ---------------------------------------------------------------------------
extern "C" void kernel_launch(void* const* d_in, const int* in_sizes, int n_in,
                              void* d_out, int out_size, void* d_ws, size_t ws_size,
                              hipStream_t stream) {
  const float*  u        = (const float*)d_in[0];   // (8, 2048, 512) f32
  const float*  Bm       = (const float*)d_in[1];   // (512, 64) f32
  const float*  Ctm      = (const float*)d_in[2];   // (512, 64) f32
  const float*  Dv       = (const float*)d_in[3];   // (512,) f32
  const float*  log_step = (const float*)d_in[4];   // (512,) f32
  const float2* p        = (const float2*)d_in[5];  // (64,) c64
  const float2* q        = (const float2*)d_in[6];  // (64,) c64
  const float2* lam      = (const float2*)d_in[7];  // (64,) c64
  float* out = (float*)d_out;                       // (8, 2048, 512) f32

  float2* Khat = (float2*)d_ws;                                               // 16 MB
  float*  ut   = (float*)((char*)d_ws + (size_t)512 * 4096 * sizeof(float2)); // 32 MB

  s4_kernel_K<<<512, 256, 0, stream>>>(Bm, Ctm, log_step, p, q, lam, Khat);
  s4_tin<<<dim3(64, 16, 8), 256, 0, stream>>>(u, ut);
  s4_conv<<<4096, 256, 0, stream>>>(ut, Khat, Dv);
  s4_tout<<<dim3(64, 16, 8), 256, 0, stream>>>(ut, out);
  (void)in_sizes; (void)n_in; (void)out_size; (void)ws_size;
}